// RGCN_60215441490198
// MI455X (gfx1250) — compile-verified
//
#include <hip/hip_runtime.h>
#include <math.h>

// ---------------- problem constants (match reference) ----------------
#define NATOMS 60000   // divisible by 16 -> no row guards in node GEMMs
#define NEDGES 132000
#define NRELS  65
#define NGRAPH 2500
#define NMOTIF 5000
#define DIN    40
#define DHID   64
#define DFFN   256
#define BN_EPS 1e-5f
#define WPB    4       // waves per block in WMMA kernels

typedef float v2f __attribute__((ext_vector_type(2)));
typedef float v8f __attribute__((ext_vector_type(8)));

__device__ __forceinline__ v8f wmma_f32_16x16x4(v2f a, v2f b, v8f c) {
    // D = A(16x4, f32) * B(4x16, f32) + C(16x16, f32)
    return __builtin_amdgcn_wmma_f32_16x16x4_f32(
        /*neg_a=*/false, a, /*neg_b=*/false, b,
        /*c_mod=*/(short)0, c, /*reuse_a=*/false, /*reuse_b=*/false);
}

// ---------------------------------------------------------------------
// Edge-typed message passing: msg[dst] += h[src] @ W[etype]   (dout=64)
// One wave per edge; lane j accumulates output cols j and j+32.
// B operand (W[etype]) is relation-divergent per edge, so this stays
// VALU; weights are L2-resident (~1MB) and atomics land in L2.
// ---------------------------------------------------------------------
__global__ void edge_msg_kernel(const float* __restrict__ h, int din,
                                const int* __restrict__ src,
                                const int* __restrict__ dst,
                                const int* __restrict__ etype,
                                const float* __restrict__ W,  // [R, din, 64]
                                float* __restrict__ msg) {    // [N, 64]
    int e = blockIdx.x * (blockDim.x >> 5) + (threadIdx.x >> 5);
    if (e >= NEDGES) return;
    int lane = threadIdx.x & 31;
    int s = src[e], d = dst[e], r = etype[e];
    const float* hs = h + (size_t)s * din;
    const float* Wr = W + (size_t)r * din * 64;
    float acc0 = 0.f, acc1 = 0.f;
    for (int k = 0; k < din; ++k) {
        float a = hs[k];                 // uniform across wave -> broadcast
        acc0 = fmaf(a, Wr[k * 64 + lane], acc0);
        acc1 = fmaf(a, Wr[k * 64 + lane + 32], acc1);
    }
    atomicAdd(&msg[(size_t)d * 64 + lane], acc0);
    atomicAdd(&msg[(size_t)d * 64 + lane + 32], acc1);
}

// ---------------------------------------------------------------------
// Fused RGCN layer GEMM + epilogue (dout = 64, NATOMS % 16 == 0):
//   out = relu(msg + h@W_loop + b) + relu(h@Wres + bres)
// One wave per 16x16 tile; shared A tile drives two WMMA chains.
// ---------------------------------------------------------------------
__global__ void rgcn_layer_kernel(const float* __restrict__ h, int din,
                                  const float* __restrict__ msg,
                                  const float* __restrict__ Wloop,  // [din,64]
                                  const float* __restrict__ bconv,  // [64]
                                  const float* __restrict__ Wres,   // [din,64]
                                  const float* __restrict__ bres,   // [64]
                                  float* __restrict__ out) {        // [N,64]
    int wave  = threadIdx.x >> 5;
    int lane  = threadIdx.x & 31;
    int tileN = blockIdx.x * WPB + wave;   // 0..3
    int tileM = blockIdx.y;                // node tile (3750 exact)
    int m   = lane & 15;
    int grp = lane >> 4;                   // 0: K=k,k+1   1: K=k+2,k+3
    const float* Arow = h + (size_t)(tileM * 16 + m) * din;
    int bcol = tileN * 16 + m;             // lane's output column

    v8f accL = {}, accR = {};
    for (int k = 0; k < din; k += 4) {
        int ka = k + grp * 2;
        v2f a = *(const v2f*)(Arow + ka);          // single b64 load
        v2f bl, br;
        bl.x = Wloop[(size_t)ka * 64 + bcol];
        bl.y = Wloop[(size_t)(ka + 1) * 64 + bcol];
        br.x = Wres[(size_t)ka * 64 + bcol];
        br.y = Wres[(size_t)(ka + 1) * 64 + bcol];
        accL = wmma_f32_16x16x4(a, bl, accL);
        accR = wmma_f32_16x16x4(a, br, accR);
    }
    float bc = bconv[bcol], brr = bres[bcol];
#pragma unroll
    for (int i = 0; i < 8; ++i) {
        int row = tileM * 16 + i + grp * 8;  // C/D layout: VGPR i -> row i (+8 upper half)
        size_t off = (size_t)row * 64 + bcol;
        float conv = fmaxf(msg[off] + accL[i] + bc, 0.f);
        float res  = fmaxf(accR[i] + brr, 0.f);
        out[off] = conv + res;
    }
}

// ---------------------------------------------------------------------
// BatchNorm over axis 0: pass 1 accumulates per-column sum / sumsq.
// Block = 256 threads = 64 cols x 4 row-lanes over 64 consecutive rows.
// ---------------------------------------------------------------------
__global__ void bn_stats_kernel(const float* __restrict__ x,
                                float* __restrict__ stats /*[128]*/,
                                int nrows) {
    __shared__ float sh[256], shq[256];
    int col = threadIdx.x & 63;
    int rl  = threadIdx.x >> 6;      // 0..3
    float s = 0.f, sq = 0.f;
    int base = blockIdx.x * 64;
    for (int i = rl; i < 64; i += 4) {
        int row = base + i;
        if (row < nrows) {
            float v = x[(size_t)row * 64 + col];
            s += v; sq += v * v;
        }
    }
    sh[threadIdx.x] = s; shq[threadIdx.x] = sq;
    __syncthreads();
    if (rl == 0) {
        float ts  = sh[col] + sh[col + 64] + sh[col + 128] + sh[col + 192];
        float tsq = shq[col] + shq[col + 64] + shq[col + 128] + shq[col + 192];
        atomicAdd(&stats[col], ts);
        atomicAdd(&stats[64 + col], tsq);
    }
}

__global__ void bn_apply_kernel(const float* __restrict__ x,
                                const float* __restrict__ stats,
                                const float* __restrict__ gamma,
                                const float* __restrict__ beta,
                                float* __restrict__ y, int nrows) {
    size_t idx = (size_t)blockIdx.x * blockDim.x + threadIdx.x;
    if (idx >= (size_t)nrows * 64) return;
    int col = (int)(idx & 63);
    float inv_n = 1.f / (float)nrows;
    float mu  = stats[col] * inv_n;
    float var = stats[64 + col] * inv_n - mu * mu;   // biased, matches torch BN
    float inv = 1.f / sqrtf(var + BN_EPS);
    y[idx] = (x[idx] - mu) * inv * gamma[col] + beta[col];
}

// ---------------------------------------------------------------------
// Atom weighting: w = sigmoid(h @ wa + ba)
// ---------------------------------------------------------------------
__global__ void atom_w_kernel(const float* __restrict__ h,
                              const float* __restrict__ wa,
                              const float* __restrict__ ba,
                              float* __restrict__ w) {
    int n = blockIdx.x * blockDim.x + threadIdx.x;
    if (n >= NATOMS) return;
    const float* hr = h + (size_t)n * 64;
    float acc = ba[0];
    for (int k = 0; k < 64; ++k) acc = fmaf(hr[k], wa[k], acc);
    w[n] = 1.f / (1.f + expf(-acc));
}

// ---------------------------------------------------------------------
// Gated segment-sum readout: graph_feats (by graph id, into d_out) and
// per-motif sums (motif 0 kept in row 0 of the buffer, dropped later).
// ---------------------------------------------------------------------
__global__ void readout_scatter_kernel(const float* __restrict__ h,
                                       const float* __restrict__ w,
                                       const float* __restrict__ smask,
                                       const float* __restrict__ smask_full,
                                       const int* __restrict__ gids,
                                       const int* __restrict__ motif,
                                       float* __restrict__ gfeats,    // [B,64]
                                       float* __restrict__ hsubs) {   // [M+1,64]
    int n = blockIdx.x;
    int c = threadIdx.x;            // 64 threads
    float wn = w[n];
    float hv = h[(size_t)n * 64 + c];
    atomicAdd(&gfeats[(size_t)gids[n] * 64 + c], hv * wn * smask[n]);
    atomicAdd(&hsubs[(size_t)motif[n] * 64 + c], hv * wn * smask_full[n]);
}

// ---------------------------------------------------------------------
// Generic WMMA GEMM: C = op(A[M,K] @ B[K,N] + bias), op = relu optional.
// One wave per 16x16 output tile, 4 waves/block. K % 4 == 0, N % 16 == 0.
// Out-of-range A rows only feed out-of-range C rows, so the clamped row
// pointer needs no zero-fill; only the store is guarded.
// ---------------------------------------------------------------------
__global__ void gemm_wmma_kernel(const float* __restrict__ A,
                                 const float* __restrict__ B,
                                 const float* __restrict__ bias,
                                 float* __restrict__ C,
                                 int Mrows, int Ncols, int K, int relu) {
    int wave  = threadIdx.x >> 5;
    int lane  = threadIdx.x & 31;
    int tileN = blockIdx.x * WPB + wave;
    int tileM = blockIdx.y;
    int m   = lane & 15;
    int grp = lane >> 4;
    int arow = tileM * 16 + m;
    const float* Arow = A + (size_t)(arow < Mrows ? arow : Mrows - 1) * K;
    int bcol = tileN * 16 + m;

    v8f acc = {};
    for (int k = 0; k < K; k += 4) {
        int ka = k + grp * 2;
        v2f a = *(const v2f*)(Arow + ka);          // single b64 load, branch-free
        v2f b;
        b.x = B[(size_t)ka * Ncols + bcol];
        b.y = B[(size_t)(ka + 1) * Ncols + bcol];
        acc = wmma_f32_16x16x4(a, b, acc);
    }
    float bv = bias ? bias[bcol] : 0.f;
#pragma unroll
    for (int i = 0; i < 8; ++i) {
        int row = tileM * 16 + i + grp * 8;
        if (row < Mrows) {
            float v = acc[i] + bv;
            if (relu) v = fmaxf(v, 0.f);
            C[(size_t)row * Ncols + bcol] = v;
        }
    }
}

// ---------------------------------------------------------------------
extern "C" void kernel_launch(void* const* d_in, const int* in_sizes, int n_in,
                              void* d_out, int out_size, void* d_ws, size_t ws_size,
                              hipStream_t stream) {
    // inputs in setup_inputs() dict order, params flattened in insertion order
    const float* node_feats = (const float*)d_in[0];   // [N,40]
    const int*   etype      = (const int*)d_in[1];
    const int*   src        = (const int*)d_in[2];
    const int*   dst        = (const int*)d_in[3];
    const float* smask      = (const float*)d_in[4];
    const float* smask_full = (const float*)d_in[5];
    const int*   motif      = (const int*)d_in[6];
    const int*   gids       = (const int*)d_in[7];
    // layer 0
    const float* W0  = (const float*)d_in[8];    // [R,40,64]
    const float* Wl0 = (const float*)d_in[9];    // [40,64]
    const float* b0  = (const float*)d_in[10];
    const float* Wr0 = (const float*)d_in[11];
    const float* br0 = (const float*)d_in[12];
    const float* g0  = (const float*)d_in[13];
    const float* be0 = (const float*)d_in[14];
    // layer 1
    const float* W1r  = (const float*)d_in[15];  // [R,64,64]
    const float* Wl1  = (const float*)d_in[16];
    const float* b1l  = (const float*)d_in[17];
    const float* Wr1  = (const float*)d_in[18];
    const float* br1  = (const float*)d_in[19];
    const float* g1   = (const float*)d_in[20];
    const float* be1  = (const float*)d_in[21];
    // heads
    const float* wa = (const float*)d_in[22];    // [64,1]
    const float* ba = (const float*)d_in[23];    // [1]
    const float* Wf = (const float*)d_in[24];    // [64,256]
    const float* bf = (const float*)d_in[25];    // [256]
    const float* W1 = (const float*)d_in[26];    // [256,256]
    const float* b1 = (const float*)d_in[27];    // [256]
    const float* W2 = (const float*)d_in[28];    // [256,128]
    const float* b2 = (const float*)d_in[29];    // [128]

    // workspace layout (floats)
    float* ws    = (float*)d_ws;
    float* msg   = ws;                                  // N*64
    float* outb  = msg + (size_t)NATOMS * 64;           // N*64
    float* hbuf  = outb + (size_t)NATOMS * 64;          // N*64
    float* stats = hbuf + (size_t)NATOMS * 64;          // 128
    float* wbuf  = stats + 128;                         // N
    float* hsubs = wbuf + NATOMS;                       // (M+1)*64
    float* fg    = hsubs + (size_t)(NMOTIF + 1) * 64;   // B*256
    float* t1    = fg + (size_t)NGRAPH * 256;           // B*256
    float* fs    = t1 + (size_t)NGRAPH * 256;           // M*256
    float* t2    = fs + (size_t)NMOTIF * 256;           // M*256

    float* gfeats_out = (float*)d_out;                              // [B,64]
    float* outg       = gfeats_out + (size_t)NGRAPH * 64;           // [B,128]
    float* outs       = outg + (size_t)NGRAPH * 128;                // [M,128]

    const int edgeBlocks = (NEDGES + 3) / 4;            // 4 waves per block
    const dim3 layerGrid(1, NATOMS / 16);               // 4 col-tiles = 4 waves in block

    // ------------------- Layer 1 (din = 40) -------------------
    hipMemsetAsync(msg, 0, (size_t)NATOMS * 64 * sizeof(float), stream);
    edge_msg_kernel<<<edgeBlocks, 128, 0, stream>>>(node_feats, DIN, src, dst, etype, W0, msg);
    rgcn_layer_kernel<<<layerGrid, 32 * WPB, 0, stream>>>(node_feats, DIN, msg, Wl0, b0, Wr0, br0, outb);
    hipMemsetAsync(stats, 0, 128 * sizeof(float), stream);
    bn_stats_kernel<<<(NATOMS + 63) / 64, 256, 0, stream>>>(outb, stats, NATOMS);
    bn_apply_kernel<<<((size_t)NATOMS * 64 + 255) / 256, 256, 0, stream>>>(outb, stats, g0, be0, hbuf, NATOMS);

    // ------------------- Layer 2 (din = 64) -------------------
    hipMemsetAsync(msg, 0, (size_t)NATOMS * 64 * sizeof(float), stream);
    edge_msg_kernel<<<edgeBlocks, 128, 0, stream>>>(hbuf, DHID, src, dst, etype, W1r, msg);
    rgcn_layer_kernel<<<layerGrid, 32 * WPB, 0, stream>>>(hbuf, DHID, msg, Wl1, b1l, Wr1, br1, outb);
    hipMemsetAsync(stats, 0, 128 * sizeof(float), stream);
    bn_stats_kernel<<<(NATOMS + 63) / 64, 256, 0, stream>>>(outb, stats, NATOMS);
    bn_apply_kernel<<<((size_t)NATOMS * 64 + 255) / 256, 256, 0, stream>>>(outb, stats, g1, be1, hbuf, NATOMS);

    // ------------------- Readout -------------------
    atom_w_kernel<<<(NATOMS + 255) / 256, 256, 0, stream>>>(hbuf, wa, ba, wbuf);
    hipMemsetAsync(gfeats_out, 0, (size_t)NGRAPH * 64 * sizeof(float), stream);
    hipMemsetAsync(hsubs, 0, (size_t)(NMOTIF + 1) * 64 * sizeof(float), stream);
    readout_scatter_kernel<<<NATOMS, 64, 0, stream>>>(hbuf, wbuf, smask, smask_full,
                                                      gids, motif, gfeats_out, hsubs);

    // ------------------- Global FFN head -------------------
    gemm_wmma_kernel<<<dim3(DFFN / 16 / WPB, (NGRAPH + 15) / 16), 32 * WPB, 0, stream>>>(
        gfeats_out, Wf, bf, fg, NGRAPH, DFFN, 64, 0);
    gemm_wmma_kernel<<<dim3(DFFN / 16 / WPB, (NGRAPH + 15) / 16), 32 * WPB, 0, stream>>>(
        fg, W1, b1, t1, NGRAPH, DFFN, DFFN, 1);
    gemm_wmma_kernel<<<dim3(128 / 16 / WPB, (NGRAPH + 15) / 16), 32 * WPB, 0, stream>>>(
        t1, W2, b2, outg, NGRAPH, 128, DFFN, 0);

    // ------------------- Motif FFN head (drop motif 0) -------------------
    gemm_wmma_kernel<<<dim3(DFFN / 16 / WPB, (NMOTIF + 15) / 16), 32 * WPB, 0, stream>>>(
        hsubs + 64, Wf, bf, fs, NMOTIF, DFFN, 64, 0);
    gemm_wmma_kernel<<<dim3(DFFN / 16 / WPB, (NMOTIF + 15) / 16), 32 * WPB, 0, stream>>>(
        fs, W1, b1, t2, NMOTIF, DFFN, DFFN, 1);
    gemm_wmma_kernel<<<dim3(128 / 16 / WPB, (NMOTIF + 15) / 16), 32 * WPB, 0, stream>>>(
        t2, W2, b2, outs, NMOTIF, 128, DFFN, 0);
}